// PointCloudFitter_66391604462138
// MI455X (gfx1250) — compile-verified
//
#include <hip/hip_runtime.h>
#include <math.h>

typedef __attribute__((ext_vector_type(2))) float v2f;
typedef __attribute__((ext_vector_type(8))) float v8f;

#define BB   4
#define NN   4096
#define MM   4096
#define EPSF 1e-8f
#define NTILES (BB * (NN / 16))   // 1024 waves / partial sums

// One wave (32 threads) per 16-row tile of transformed source points.
// Scans all M targets in 16-col tiles with V_WMMA_F32_16X16X4_F32:
//   A row  = [-2Tx, -2Ty, -2Tz, 1]
//   B col  = [ qx,   qy,   qz, ||q||^2 ]
//   D      = -2 T.q + ||q||^2   (add ||T||^2 after the min)
__global__ __launch_bounds__(32) void pcf_tile_kernel(
    const float* __restrict__ src, const float* __restrict__ tgt,
    const float* __restrict__ rot, const float* __restrict__ trans,
    float* __restrict__ out_transformed, float* __restrict__ ws_partial)
{
    const int lane  = threadIdx.x;      // 0..31
    const int tile  = blockIdx.x;       // 0..NTILES-1
    const int b     = tile >> 8;        // / (NN/16)
    const int ntile = tile & 255;
    const int row   = lane & 15;        // source row within tile / target col within tile
    const int hi    = lane >> 4;        // 0: lanes 0-15 (K=0,1), 1: lanes 16-31 (K=2,3)

    // ---- Rodrigues exp map (uniform across wave) ----
    const float wx = rot[0], wy = rot[1], wz = rot[2];
    const float th2 = wx * wx + wy * wy + wz * wz;
    const float th  = sqrtf(th2 + EPSF);
    const float Af  = sinf(th) / th;
    const float Bf  = (1.0f - cosf(th)) / (th2 + EPSF);
    const float tx = trans[0], ty = trans[1], tz = trans[2];
    const float R00 = 1.0f - Bf * (wy * wy + wz * wz);
    const float R01 = -Af * wz + Bf * wx * wy;
    const float R02 =  Af * wy + Bf * wx * wz;
    const float R10 =  Af * wz + Bf * wx * wy;
    const float R11 = 1.0f - Bf * (wx * wx + wz * wz);
    const float R12 = -Af * wx + Bf * wy * wz;
    const float R20 = -Af * wy + Bf * wx * wz;
    const float R21 =  Af * wx + Bf * wy * wz;
    const float R22 = 1.0f - Bf * (wx * wx + wy * wy);

    // ---- Load + transform this wave's 16 source points (each row held twice) ----
    const int n = ntile * 16 + row;
    const float* sp = src + (size_t)(b * NN + n) * 3;
    const float px = sp[0], py = sp[1], pz = sp[2];
    const float Tx = R00 * px + R01 * py + R02 * pz + tx;
    const float Ty = R10 * px + R11 * py + R12 * pz + ty;
    const float Tz = R20 * px + R21 * py + R22 * pz + tz;
    const float x2 = Tx * Tx + Ty * Ty + Tz * Tz;

    if (hi == 0) {   // write transformed output once per row
        float* op = out_transformed + (size_t)(b * NN + n) * 3;
        op[0] = Tx; op[1] = Ty; op[2] = Tz;
    }

    // ---- A operand (16x4 f32 WMMA layout) ----
    v2f a;
    a.x = hi ? (-2.0f * Tz) : (-2.0f * Tx);   // K=2 : K=0
    a.y = hi ? 1.0f         : (-2.0f * Ty);   // K=3 : K=1

    v8f cmin;
    #pragma unroll
    for (int r = 0; r < 8; ++r) cmin[r] = 3.0e38f;

    const float* tbase = tgt + (size_t)b * MM * 3;

    // ---- Scan all target tiles: WMMA + running elementwise min ----
    #pragma unroll 4
    for (int mt = 0; mt < MM / 16; ++mt) {
        const float* tp = tbase + (size_t)(mt * 16 + row) * 3;
        const float qx = tp[0], qy = tp[1], qz = tp[2];
        const float y2 = qx * qx + qy * qy + qz * qz;
        v2f bop;
        bop.x = hi ? qz : qx;                 // K=2 : K=0
        bop.y = hi ? y2 : qy;                 // K=3 : K=1

        v8f c = {};                           // fresh tile: C = 0
        c = __builtin_amdgcn_wmma_f32_16x16x4_f32(
                /*neg_a=*/false, a, /*neg_b=*/false, bop,
                /*c_mod=*/(short)0, c, /*reuse_a=*/false, /*reuse_b=*/false);

        #pragma unroll
        for (int r = 0; r < 8; ++r) cmin[r] = fminf(cmin[r], c[r]);
    }

    // ---- Min across the 16 columns (stays within each 16-lane half) ----
    #pragma unroll
    for (int r = 0; r < 8; ++r) {
        float v = cmin[r];
        v = fminf(v, __shfl_xor(v, 1, 32));
        v = fminf(v, __shfl_xor(v, 2, 32));
        v = fminf(v, __shfl_xor(v, 4, 32));
        v = fminf(v, __shfl_xor(v, 8, 32));
        cmin[r] = v;
    }

    // ---- Add ||T||^2 per row and accumulate the loss partial ----
    float s = 0.0f;
    #pragma unroll
    for (int r = 0; r < 8; ++r) {
        const int srcRow = hi ? (r + 8) : r;      // lane holding x2 for this D row
        s += cmin[r] + __shfl(x2, srcRow, 32);
    }
    s += __shfl_xor(s, 16, 32);                    // rows 0-7 half + rows 8-15 half
    if (lane == 0) ws_partial[tile] = s;
}

// Deterministic final reduction of the 1024 per-tile partials -> mean loss.
__global__ __launch_bounds__(32) void pcf_reduce_kernel(
    const float* __restrict__ ws_partial, float* __restrict__ out_loss)
{
    const int lane = threadIdx.x;
    float s = 0.0f;
    for (int i = lane; i < NTILES; i += 32) s += ws_partial[i];
    #pragma unroll
    for (int m = 16; m >= 1; m >>= 1) s += __shfl_xor(s, m, 32);
    if (lane == 0) out_loss[0] = s * (1.0f / (float)(BB * NN));
}

extern "C" void kernel_launch(void* const* d_in, const int* in_sizes, int n_in,
                              void* d_out, int out_size, void* d_ws, size_t ws_size,
                              hipStream_t stream)
{
    const float* src   = (const float*)d_in[0];   // [4,4096,3]
    const float* tgt   = (const float*)d_in[1];   // [4,4096,3]
    const float* rot   = (const float*)d_in[2];   // [3]
    const float* trans = (const float*)d_in[3];   // [3]
    float* out = (float*)d_out;                   // transformed (49152) + loss (1)
    float* ws  = (float*)d_ws;                    // 1024 partial sums

    pcf_tile_kernel<<<dim3(NTILES), dim3(32), 0, stream>>>(
        src, tgt, rot, trans, out, ws);
    pcf_reduce_kernel<<<dim3(1), dim3(32), 0, stream>>>(
        ws, out + (size_t)BB * NN * 3);
}